// BilateralFilter_25271587570278
// MI455X (gfx1250) — compile-verified
//
#include <hip/hip_runtime.h>
#include <math.h>

#ifndef __has_builtin
#define __has_builtin(x) 0
#endif

// ---- tile geometry -------------------------------------------------------
#define TZ 4                     // tile rows (Z)
#define TA 64                    // tile cols (A)  -> 256 threads = 1 px/thread
#define HZ (TZ + 2)              // halo rows   (3x5 window -> +-1 in Z)
#define HA (TA + 4)              //             (+-2 in A)
#define HROW (HA * 3)            // floats per halo row = 204
#define HTOT (HZ * HROW)         // 1224 floats = 4896 B of LDS

#define NTHREADS 256
#define K_NBR 14

// exp(-d/(2*theta^2)) = exp2(d * K),  K = -log2(e)/(2*theta^2)
// theta = {0.015, 0.015, 0.01, 0.01} -> only two distinct factors
static constexpr float KC0 = (float)(-1.4426950408889634 / (2.0 * 0.015 * 0.015));
static constexpr float KC2 = (float)(-1.4426950408889634 / (2.0 * 0.010 * 0.010));

// Address-space-qualified pointer types for the async builtin
typedef __attribute__((address_space(1))) int as1_int;   // global
typedef __attribute__((address_space(3))) int as3_int;   // LDS

// ---- CDNA5 async global->LDS copy (ASYNCcnt path), with fallbacks --------
__device__ __forceinline__ void async_copy_b32(const float* g, float* l) {
#if __has_builtin(__builtin_amdgcn_global_load_async_to_lds_b32)
  __builtin_amdgcn_global_load_async_to_lds_b32(
      (as1_int*)g, (as3_int*)l, /*offset=*/0, /*cpol=*/0);
#else
  *l = *g;  // plain fallback (still correct, just not async)
#endif
}

__device__ __forceinline__ void wait_async0() {
#if __has_builtin(__builtin_amdgcn_s_wait_asynccnt)
  __builtin_amdgcn_s_wait_asynccnt(0);
#else
  asm volatile("s_wait_asynccnt 0" ::: "memory");
#endif
}

__device__ __forceinline__ float fast_exp2(float x) {
#if __has_builtin(__builtin_amdgcn_exp2f)
  return __builtin_amdgcn_exp2f(x);   // v_exp_f32
#else
  return exp2f(x);                    // HIP device overload (host pass safe)
#endif
}

// ---- kernel --------------------------------------------------------------
__global__ __launch_bounds__(NTHREADS)
void bilateral_weights_kernel(const float* __restrict__ x,
                              float4* __restrict__ out,
                              int Z, int A) {
  __shared__ float lds[HTOT];

  const int t  = threadIdx.x;
  const int a0 = blockIdx.x * TA;      // tile origin along A
  const int z0 = blockIdx.y * TZ;      // tile origin along Z
  const int b  = blockIdx.z;           // batch

  // 1) zero-fill halo (implements jnp.pad zeros at image borders)
  #pragma unroll
  for (int i = t; i < HTOT; i += NTHREADS) lds[i] = 0.0f;
  __syncthreads();

  // 2) async-load the in-bounds halo floats (rows are contiguous in memory)
  const long rowStrideF = (long)A * 3;               // floats per (b,z) row
  for (int i = t; i < HTOT; i += NTHREADS) {
    const int r  = i / HROW;
    const int c  = i - r * HROW;
    const int z  = z0 - 1 + r;
    const int fa = a0 * 3 - 6 + c;                   // float index in row
    if (z >= 0 && z < Z && fa >= 0 && fa < A * 3) {
      const float* g = x + (long)(b * Z + z) * rowStrideF + fa;
      async_copy_b32(g, &lds[i]);
    }
  }
  wait_async0();
  __syncthreads();

  // 3) per-pixel stencil from LDS
  const int la = t & (TA - 1);
  const int lz = t >> 6;

  const int cb = (lz + 1) * HROW + (la + 2) * 3;     // center pixel in halo
  const float cx = lds[cb + 0];
  const float cy = lds[cb + 1];
  const float cz = lds[cb + 2];

  const long pix = (long)(b * Z + z0 + lz) * A + (a0 + la);
  float4* __restrict__ o = out + pix * K_NBR;        // 14 float4 = 224 B / px

  int k = 0;
  #pragma unroll
  for (int i = 0; i < 3; ++i) {
    #pragma unroll
    for (int j = 0; j < 5; ++j) {
      if (i == 1 && j == 2) continue;                // skip center
      const int nb = (lz + i) * HROW + (la + j) * 3;
      const float dx = cx - lds[nb + 0];
      const float dy = cy - lds[nb + 1];
      const float dz = cz - lds[nb + 2];
      const float d  = fmaf(dx, dx, fmaf(dy, dy, dz * dz));
      const float e0 = fast_exp2(d * KC0);           // classes 0,1 (theta=.015)
      const float e2 = fast_exp2(d * KC2);           // classes 2,3 (theta=.01)
      o[k++] = make_float4(e0, e0, e2, e2);          // global_store_b128
    }
  }
}

// ---- launcher ------------------------------------------------------------
extern "C" void kernel_launch(void* const* d_in, const int* in_sizes, int n_in,
                              void* d_out, int out_size, void* d_ws, size_t ws_size,
                              hipStream_t stream) {
  (void)n_in; (void)out_size; (void)d_ws; (void)ws_size;

  const float* x = (const float*)d_in[0];
  float4* out    = (float4*)d_out;

  const int Z = 64, A = 512;
  const int B = in_sizes[0] / (Z * A * 3);           // = 16

  dim3 grid(A / TA, Z / TZ, B);                      // 8 x 16 x 16 = 2048 WGs
  dim3 block(NTHREADS);                              // 8 wave32s
  hipLaunchKernelGGL(bilateral_weights_kernel, grid, block, 0, stream,
                     x, out, Z, A);
}